// Bayer_8907762172168
// MI455X (gfx1250) — compile-verified
//
#include <hip/hip_runtime.h>

// Bayer mosaic channel-select gather.
// img: [B=8, C=3, H=2048, W=2048] f32, out: [B, H, W] f32
// c(i,j) = 1 if (i+j) even; else 2 if i even; else 0.
// Per row the output alternates between two channel planes:
//   even row i: even j -> ch1, odd j -> ch2
//   odd  row i: even j -> ch0, odd j -> ch1
// => channel at even j = 1 - (i&1), channel at odd j = 2 - (i&1).

typedef float f4 __attribute__((ext_vector_type(4)));

// Fixed problem shape from the reference.
constexpr int    kB = 8;
constexpr int    kH = 2048;
constexpr int    kW = 2048;
constexpr size_t kPlane    = (size_t)kH * kW;   // 4,194,304 elements
constexpr size_t kChStride = 3 * kPlane;        // per-batch stride

__global__ __launch_bounds__(256) void bayer_select_kernel(
    const float* __restrict__ img, float* __restrict__ out)
{
    // Each thread handles one aligned quad of 4 consecutive output pixels.
    size_t q   = (size_t)blockIdx.x * blockDim.x + threadIdx.x; // quad index
    // W/4 = 512 quads per row; H = 2048 rows per batch.
    size_t row = q >> 9;                 // global row = b*H + i
    int    jq  = (int)(q & 511);         // quad index within the row
    int    i   = (int)(row & (size_t)(kH - 1));
    size_t b   = row >> 11;

    int iOdd = i & 1;
    size_t off = ((size_t)i << 11) + ((size_t)jq << 2); // i*W + 4*jq

    const float* base = img + b * kChStride;
    const f4* pEven = (const f4*)(base + (size_t)(1 - iOdd) * kPlane + off); // channel for even j
    const f4* pOdd  = (const f4*)(base + (size_t)(2 - iOdd) * kPlane + off); // channel for odd j

    // Streaming data (no reuse, footprint >> L2): non-temporal loads/stores.
    f4 a = __builtin_nontemporal_load(pEven);
    f4 c = __builtin_nontemporal_load(pOdd);

    f4 r;
    r.x = a.x;  // j0   (even) from pEven
    r.y = c.y;  // j0+1 (odd)  from pOdd
    r.z = a.z;  // j0+2 (even) from pEven
    r.w = c.w;  // j0+3 (odd)  from pOdd

    __builtin_nontemporal_store(r, (f4*)(out + (q << 2)));
}

extern "C" void kernel_launch(void* const* d_in, const int* in_sizes, int n_in,
                              void* d_out, int out_size, void* d_ws, size_t ws_size,
                              hipStream_t stream) {
    (void)in_sizes; (void)n_in; (void)d_ws; (void)ws_size; (void)out_size;
    const float* img = (const float*)d_in[0];
    float*       out = (float*)d_out;

    const size_t quads   = (size_t)kB * kH * kW / 4;  // 8,388,608 threads
    const int    block   = 256;                       // 8 wave32 waves
    const unsigned grid  = (unsigned)(quads / block); // 32,768 blocks

    bayer_select_kernel<<<grid, block, 0, stream>>>(img, out);
}